// EMA_VQ_23536420782581
// MI455X (gfx1250) — compile-verified
//
#include <hip/hip_runtime.h>
#include <math.h>

#define N_PTS   262144
#define D_HALF  64
#define D2      128
#define KCODES  512
#define BETA    0.25f
#define DECAY   0.99f
#define EPSV    1e-5f
#define EMB_STRIDE 132   // padded LDS row stride (floats): bank stride 4 -> conflict-free b64 reads

typedef __attribute__((ext_vector_type(2))) float v2f;
typedef __attribute__((ext_vector_type(8))) float v8f;

__global__ void vq_zero_ws(float* ws, int n) {
    int i = blockIdx.x * blockDim.x + threadIdx.x;
    if (i < n) ws[i] = 0.0f;
}

__global__ __launch_bounds__(256) void vq_main(
    const float* __restrict__ z_real, const float* __restrict__ z_imag,
    const float* __restrict__ emb_w,
    float* __restrict__ out_zq_real, float* __restrict__ out_zq_imag,
    float* __restrict__ out_loss, float* __restrict__ out_idx,
    float* __restrict__ ws_ntotal, float* __restrict__ ws_dw)
{
    extern __shared__ float smem[];
    float* s_emb = smem;                          // KCODES * EMB_STRIDE
    float* s_ysq = smem + KCODES * EMB_STRIDE;    // KCODES

    const int tid = threadIdx.x;

    // Stage full codebook (256 KB) into padded LDS, cooperatively, float4 loads.
    for (int t = tid; t < KCODES * 32; t += 256) {
        const int code = t >> 5;
        const int seg  = (t & 31) << 2;
        const float4 v = *(const float4*)(emb_w + code * D2 + seg);
        float* d = s_emb + code * EMB_STRIDE + seg;
        d[0] = v.x; d[1] = v.y; d[2] = v.z; d[3] = v.w;
    }
    __syncthreads();
    // Per-code squared norms.
    for (int code = tid; code < KCODES; code += 256) {
        const float* r = s_emb + code * EMB_STRIDE;
        float s = 0.f;
        #pragma unroll 8
        for (int j = 0; j < D2; ++j) s += r[j] * r[j];
        s_ysq[code] = s;
    }
    __syncthreads();

    const int lane   = tid & 31;
    const int waveId = tid >> 5;
    const int m      = lane & 15;                // row within strip (A) / column within tile (B,C)
    const int koff   = (lane < 16) ? 0 : 2;      // ISA 16x4 fp32 A/B lane layout

    for (int s = 0; s < 4; ++s) {
        const int strip = blockIdx.x * 32 + waveId * 4 + s;   // 512*32 = 16384 strips of 16 rows
        const int row0  = strip * 16;
        const int rowA  = row0 + m;

        // A tile: 16 rows x 128 k of z_flat in 64 VGPRs (v2f per k-step).
        v2f a[32];
        #pragma unroll
        for (int ks = 0; ks < 32; ++ks) {
            const int k = ks * 4 + koff;
            const float* src = (k < D_HALF) ? (z_real + rowA * D_HALF + k)
                                            : (z_imag + rowA * D_HALF + (k - D_HALF));
            a[ks].x = src[0];
            a[ks].y = src[1];
        }

        float minval[8];
        int   minidx[8];
        #pragma unroll
        for (int r = 0; r < 8; ++r) { minval[r] = 3.4e38f; minidx[r] = 0; }

        // 16 iterations x (2 column tiles, 2 independent accumulator chains) x 32 k-steps.
        // Dual accumulators break the serial WMMA D->C dependence chain (distance 2),
        // keeping the matrix pipe busy at our low (LDS-limited) occupancy.
        for (int tile = 0; tile < 32; tile += 2) {
            const int n0 = tile * 16 + m;
            const int n1 = n0 + 16;
            const float* bp0 = s_emb + n0 * EMB_STRIDE + koff;
            const float* bp1 = s_emb + n1 * EMB_STRIDE + koff;
            v8f acc0 = {};
            v8f acc1 = {};
            #pragma unroll
            for (int ks = 0; ks < 32; ++ks) {
                const v2f b0 = *(const v2f*)(bp0 + ks * 4);
                const v2f b1 = *(const v2f*)(bp1 + ks * 4);
                acc0 = __builtin_amdgcn_wmma_f32_16x16x4_f32(
                           false, a[ks], false, b0, (short)0, acc0, false, false);
                acc1 = __builtin_amdgcn_wmma_f32_16x16x4_f32(
                           false, a[ks], false, b1, (short)0, acc1, false, false);
            }
            const float ysq0 = s_ysq[n0];
            const float ysq1 = s_ysq[n1];
            #pragma unroll
            for (int r = 0; r < 8; ++r) {
                const float v0 = ysq0 - 2.0f * acc0[r];  // argmin invariant to per-row x^2
                if (v0 < minval[r]) { minval[r] = v0; minidx[r] = n0; }
                const float v1 = ysq1 - 2.0f * acc1[r];
                if (v1 < minval[r]) { minval[r] = v1; minidx[r] = n1; }
            }
        }

        // Cross-lane argmin within each 16-lane half (rows 0-7 low half, 8-15 high half).
        #pragma unroll
        for (int r = 0; r < 8; ++r) {
            #pragma unroll
            for (int msk = 1; msk <= 8; msk <<= 1) {
                const float ov = __shfl_xor(minval[r], msk, 32);
                const int   oi = __shfl_xor(minidx[r], msk, 32);
                if (ov < minval[r] || (ov == minval[r] && oi < minidx[r])) {
                    minval[r] = ov; minidx[r] = oi;
                }
            }
        }

        // Gather z_q, write outputs, segment-sum atomics. Fully coalesced: lane -> 4 dims.
        const int c = lane * 4;   // 0..124
        #pragma unroll
        for (int rr = 0; rr < 16; ++rr) {
            const int srcLane = (rr < 8) ? 0 : 16;
            const int bi  = __shfl(minidx[rr & 7], srcLane, 32);
            const int row = row0 + rr;

            const float* q = s_emb + bi * EMB_STRIDE + c;
            const float qx = q[0], qy = q[1], qz = q[2], qw = q[3];

            const float* zsrc = (lane < 16) ? (z_real + row * D_HALF + c)
                                            : (z_imag + row * D_HALF + (c - D_HALF));
            const float4 zv = *(const float4*)zsrc;

            float* qdst = (lane < 16) ? (out_zq_real + row * D_HALF + c)
                                      : (out_zq_imag + row * D_HALF + (c - D_HALF));
            float4 qo; qo.x = qx; qo.y = qy; qo.z = qz; qo.w = qw;
            *(float4*)qdst = qo;

            const float dx = qx - zv.x, dy = qy - zv.y, dz = qz - zv.z, dwv = qw - zv.w;
            float dsq = dx * dx + dy * dy + dz * dz + dwv * dwv;
            #pragma unroll
            for (int msk = 16; msk >= 1; msk >>= 1) dsq += __shfl_xor(dsq, msk, 32);

            atomicAdd(&ws_dw[bi * D2 + c + 0], zv.x);
            atomicAdd(&ws_dw[bi * D2 + c + 1], zv.y);
            atomicAdd(&ws_dw[bi * D2 + c + 2], zv.z);
            atomicAdd(&ws_dw[bi * D2 + c + 3], zv.w);
            if (lane == 0) {
                out_loss[row] = BETA * dsq * (1.0f / 128.0f);
                out_idx[row]  = (float)bi;
                atomicAdd(&ws_ntotal[bi], 1.0f);
            }
        }
    }
}

__global__ __launch_bounds__(512) void vq_finalize(
    const float* __restrict__ ema_cluster_size, const float* __restrict__ ema_w,
    const float* __restrict__ ws_ntotal, const float* __restrict__ ws_dw,
    float* __restrict__ out_entropy, float* __restrict__ out_new_emb_w,
    float* __restrict__ out_new_cluster, float* __restrict__ out_new_ema_w)
{
    __shared__ float red[512];
    const int k = threadIdx.x;

    const float nt  = ws_ntotal[k];
    const float ncl = ema_cluster_size[k] * DECAY + (1.0f - DECAY) * nt;
    out_new_cluster[k] = ncl;

    red[k] = ncl;
    __syncthreads();
    for (int off = 256; off > 0; off >>= 1) {
        if (k < off) red[k] += red[k + off];
        __syncthreads();
    }
    const float tot = red[0];
    __syncthreads();

    const float csize = (ncl + EPSV) / (tot + (float)KCODES * EPSV) * tot;
    const float inv = 1.0f / csize;
    for (int c = 0; c < D2; ++c) {
        const float ne = ema_w[k * D2 + c] * DECAY + (1.0f - DECAY) * ws_dw[k * D2 + c];
        out_new_ema_w[k * D2 + c] = ne;
        out_new_emb_w[k * D2 + c] = ne * inv;
    }

    const float p = nt / (float)N_PTS;
    red[k] = -(p * logf(p + 1e-10f));
    __syncthreads();
    for (int off = 256; off > 0; off >>= 1) {
        if (k < off) red[k] += red[k + off];
        __syncthreads();
    }
    if (k == 0) out_entropy[0] = red[0] / logf((float)KCODES);
}

extern "C" void kernel_launch(void* const* d_in, const int* in_sizes, int n_in,
                              void* d_out, int out_size, void* d_ws, size_t ws_size,
                              hipStream_t stream) {
    const float* z_real = (const float*)d_in[0];
    const float* z_imag = (const float*)d_in[1];
    const float* emb_w  = (const float*)d_in[2];
    const float* ema_cs = (const float*)d_in[3];
    const float* ema_w  = (const float*)d_in[4];

    float* out          = (float*)d_out;
    float* o_zq_real    = out;
    float* o_zq_imag    = o_zq_real + (size_t)N_PTS * D_HALF;
    float* o_loss       = o_zq_imag + (size_t)N_PTS * D_HALF;
    float* o_idx        = o_loss + N_PTS;
    float* o_ent        = o_idx + N_PTS;
    float* o_new_emb    = o_ent + 1;
    float* o_new_clust  = o_new_emb + KCODES * D2;
    float* o_new_ema    = o_new_clust + KCODES;

    float* ws    = (float*)d_ws;
    float* ws_nt = ws;               // KCODES
    float* ws_dw = ws + KCODES;      // KCODES * D2

    const int zn = KCODES + KCODES * D2;
    vq_zero_ws<<<(zn + 255) / 256, 256, 0, stream>>>(ws, zn);

    const size_t smem = (size_t)(KCODES * EMB_STRIDE + KCODES) * sizeof(float);
    hipFuncSetAttribute(reinterpret_cast<const void*>(vq_main),
                        hipFuncAttributeMaxDynamicSharedMemorySize, (int)smem);
    vq_main<<<512, 256, smem, stream>>>(z_real, z_imag, emb_w,
                                        o_zq_real, o_zq_imag, o_loss, o_idx,
                                        ws_nt, ws_dw);

    vq_finalize<<<1, 512, 0, stream>>>(ema_cs, ema_w, ws_nt, ws_dw,
                                       o_ent, o_new_emb, o_new_clust, o_new_ema);
}